// MultiHeadBase_3710851744298
// MI455X (gfx1250) — compile-verified
//
#include <hip/hip_runtime.h>

// MI455X / gfx1250 attention (compile-only target).
// reference: scores = QK^T/8 ; log_softmax(k-axis) ; where(mask==0,-1e9) ; @V
//
// fp32 WMMA throughout (problem is L2-bandwidth bound, not compute bound).
// CDNA5 paths used: v_wmma_f32_16x16x4_f32, global_load_async_to_lds_b128
// double-buffered staging with s_wait_asynccnt, split workgroup barriers,
// per-lane online logsumexp (no per-tile ds_bpermute traffic).

typedef float v2f __attribute__((ext_vector_type(2)));
typedef float v8f __attribute__((ext_vector_type(8)));

#define B_   4
#define H_   16
#define S_   2048
#define D_   64
#define QT_  128               // q rows per workgroup = 8 waves x 16
#define KT_  32                // k columns per staged tile (2 WMMA subtiles)
#define NST_ (S_ / KT_)        // 64 stages
#define KROW_ 68               // padded LDS row stride (floats): conflict-free frags
#define SCALE 0.125f           // 1/sqrt(64)
#define NEGINF (-1e9f)

#define WAIT_ASYNC0() asm volatile("s_wait_asynccnt 0x0" ::: "memory")
#define WAIT_ASYNC2() asm volatile("s_wait_asynccnt 0x2" ::: "memory")
#define WAIT_ASYNC4() asm volatile("s_wait_asynccnt 0x4" ::: "memory")

// Async-DMA one 32x64 f32 tile (8KB) global -> padded LDS rows.
// 256 threads x 2 issues x 16B = 8KB. Tracked on ASYNCcnt (2 per wave).
__device__ __forceinline__ void stage_tile(const float* __restrict__ gsrc,
                                           float* lds_base, int tid) {
#pragma unroll
  for (int it = 0; it < 2; ++it) {
    const int c    = it * 256 + tid;        // 16B chunk id (512 chunks)
    const int row  = c >> 4;                // 16 chunks per 256B source row
    const int colb = (c & 15) << 4;         // byte offset within row
    const unsigned lb =
        (unsigned)(uintptr_t)lds_base + (unsigned)(row * (KROW_ * 4) + colb);
    const float* g = gsrc + row * D_ + (colb >> 2);
    asm volatile("global_load_async_to_lds_b128 %0, %1, off"
                 :: "v"(lb), "v"(g) : "memory");
  }
}

__global__ __launch_bounds__(256)
void mha_logsoftmax_postmask_kernel(const float* __restrict__ Q,
                                    const float* __restrict__ K,
                                    const float* __restrict__ V,
                                    const int*   __restrict__ mask,
                                    float*       __restrict__ out) {
  __shared__ float kbuf[2][KT_][KROW_];   // double-buffered staged K tile
  __shared__ float vbuf[2][KT_][KROW_];   // double-buffered staged V tile
  __shared__ float lds_p[8][16][18];      // per-wave P transpose slab

  const int tid  = threadIdx.x;
  const int lane = tid & 31;
  const int wave = tid >> 5;              // 8 waves
  const int half = lane >> 4;             // wave32 halves
  const int ln   = lane & 15;

  const int bh = blockIdx.y;
  const int q0 = blockIdx.x * QT_ + wave * 16;   // this wave's 16 q rows

  const float* Qb = Q + ((size_t)bh * S_ + q0) * D_;
  const float* Kb = K + (size_t)bh * S_ * D_;
  const float* Vb = V + (size_t)bh * S_ * D_;

  // ---- Q fragments, loaded once (A-matrix 16x4 f32 layout), reused twice.
  v2f aq[16];
  {
    const float* qrow = Qb + ln * D_ + 2 * half;
#pragma unroll
    for (int j = 0; j < 16; ++j)
      aq[j] = *(const v2f*)(qrow + 4 * j);
  }

  // ============== pass 1: logsumexp over all k (mask is post-softmax) ======
  // Per-lane online (m,l): each lane sees only columns == ln (mod 16) of its
  // 8 rows, so no cross-lane work inside the loop; one butterfly merge after.
  float mrow[8], lrow[8];
#pragma unroll
  for (int r = 0; r < 8; ++r) { mrow[r] = -INFINITY; lrow[r] = 0.0f; }

  stage_tile(Kb, &kbuf[0][0][0], tid);
  for (int kt = 0; kt < NST_; ++kt) {
    const int cur = kt & 1;
    if (kt + 1 < NST_) {
      stage_tile(Kb + (size_t)(kt + 1) * KT_ * D_, &kbuf[cur ^ 1][0][0], tid);
      WAIT_ASYNC2();                        // our 2 stores for tile kt landed
    } else {
      WAIT_ASYNC0();
    }
    __syncthreads();                        // tile kt visible to all waves

#pragma unroll
    for (int s = 0; s < 2; ++s) {           // two 16-wide subtiles
      v8f sacc = {};
#pragma unroll
      for (int j = 0; j < 16; ++j) {
        v2f bk = *(const v2f*)&kbuf[cur][s * 16 + ln][4 * j + 2 * half];
        sacc = __builtin_amdgcn_wmma_f32_16x16x4_f32(
            false, aq[j], false, bk, (short)0, sacc, false, false);
      }
#pragma unroll
      for (int r = 0; r < 8; ++r) {
        float t  = sacc[r] * SCALE;
        float mn = fmaxf(mrow[r], t);
        lrow[r]  = lrow[r] * __expf(mrow[r] - mn) + __expf(t - mn);
        mrow[r]  = mn;
      }
    }
    __syncthreads();                        // all waves done reading buf[cur]
  }

  // one-time 16-lane logsumexp butterfly merge (stays within each half)
  float lse[8];
#pragma unroll
  for (int r = 0; r < 8; ++r) {
    float m = mrow[r], l = lrow[r];
#pragma unroll
    for (int sft = 1; sft <= 8; sft <<= 1) {
      float om = __shfl_xor(m, sft);
      float ol = __shfl_xor(l, sft);
      float mn = fmaxf(m, om);
      l = l * __expf(m - mn) + ol * __expf(om - mn);
      m = mn;
    }
    lse[r] = m + __logf(l);                 // row (r + 8*half)
  }

  // ============== pass 2: P = logsoftmax -> post-mask ; O += P·V ===========
  v8f oacc[4];
#pragma unroll
  for (int d = 0; d < 4; ++d) oacc[d] = (v8f){};

  const int* mrowp[8];
#pragma unroll
  for (int r = 0; r < 8; ++r)
    mrowp[r] = mask + (size_t)(q0 + r + 8 * half) * S_ + ln;

  float (*pl)[18] = lds_p[wave];

  stage_tile(Kb, &kbuf[0][0][0], tid);
  stage_tile(Vb, &vbuf[0][0][0], tid);
  for (int kt = 0; kt < NST_; ++kt) {
    const int cur = kt & 1;
    if (kt + 1 < NST_) {
      const size_t off = (size_t)(kt + 1) * KT_ * D_;
      stage_tile(Kb + off, &kbuf[cur ^ 1][0][0], tid);
      stage_tile(Vb + off, &vbuf[cur ^ 1][0][0], tid);
      WAIT_ASYNC4();
    } else {
      WAIT_ASYNC0();
    }
    __syncthreads();

#pragma unroll
    for (int s = 0; s < 2; ++s) {
      const int kcol = kt * KT_ + s * 16;
      // --- scores for this 16x16 subtile
      v8f sacc = {};
#pragma unroll
      for (int j = 0; j < 16; ++j) {
        v2f bk = *(const v2f*)&kbuf[cur][s * 16 + ln][4 * j + 2 * half];
        sacc = __builtin_amdgcn_wmma_f32_16x16x4_f32(
            false, aq[j], false, bk, (short)0, sacc, false, false);
      }
      // --- log-softmax, then post-softmax masking; C-layout -> LDS slab
#pragma unroll
      for (int r = 0; r < 8; ++r) {
        float p = sacc[r] * SCALE - lse[r];
        int  mv = mrowp[r][kcol];
        p = (mv == 0) ? NEGINF : p;
        pl[r + 8 * half][ln] = p;           // (M = r+8*half, N = ln)
      }
      // same-wave DS ordering (DScnt, in-order) covers store->load

      // --- O += P(16x16) · V(16x64), A-frags re-read transposed from LDS
      const float* prow = &pl[ln][2 * half];
#pragma unroll
      for (int j = 0; j < 4; ++j) {
        v2f ap = *(const v2f*)(prow + 4 * j);       // ds_load_b64
        const int vr = s * 16 + 4 * j + 2 * half;   // V row (k index)
#pragma unroll
        for (int dsub = 0; dsub < 4; ++dsub) {
          v2f bv;
          bv.x = vbuf[cur][vr][dsub * 16 + ln];
          bv.y = vbuf[cur][vr + 1][dsub * 16 + ln];
          oacc[dsub] = __builtin_amdgcn_wmma_f32_16x16x4_f32(
              false, ap, false, bv, (short)0, oacc[dsub], false, false);
        }
      }
    }
    __syncthreads();
  }

  // ============== epilogue: (B,H,Sq,D) -> (B,Sq,H*D) =======================
  const int b = bh / H_;
  const int h = bh % H_;
#pragma unroll
  for (int dsub = 0; dsub < 4; ++dsub) {
#pragma unroll
    for (int r = 0; r < 8; ++r) {
      const int qrow = q0 + r + 8 * half;
      out[((size_t)b * S_ + qrow) * (H_ * D_) + h * D_ + dsub * 16 + ln] =
          oacc[dsub][r];
    }
  }
}

extern "C" void kernel_launch(void* const* d_in, const int* in_sizes, int n_in,
                              void* d_out, int out_size, void* d_ws, size_t ws_size,
                              hipStream_t stream) {
  const float* q    = (const float*)d_in[0];
  const float* k    = (const float*)d_in[1];
  const float* v    = (const float*)d_in[2];
  const int*   mask = (const int*)d_in[3];
  float* out = (float*)d_out;

  dim3 grid(S_ / QT_, B_ * H_);   // 16 q-tiles x 64 (b,h) = 1024 workgroups
  mha_logsoftmax_postmask_kernel<<<grid, 256, 0, stream>>>(q, k, v, mask, out);
}